// KMeans_70282844832088
// MI455X (gfx1250) — compile-verified
//
#include <hip/hip_runtime.h>
#include <math.h>

// ---------------- problem constants ----------------
#define N_PTS   65536
#define NCLU    256
#define NFEAT   256
#define BZ      2
#define MAX_IT  20
#define TOLV    1e-4f
#define TOPK    15
#define NEGV    (-1e30f)

// assign kernel geometry: 128 blocks * 8 waves * 4 tiles * 16 pts = 65536
#define NBLK    128
#define TPW     4

typedef __attribute__((ext_vector_type(2))) float v2f;
typedef __attribute__((ext_vector_type(8))) float v8f;

// ---------------------------------------------------------------------------
// init: per-batch reset. cents kept padded as [256][4]; npts=1; active=1.
// ---------------------------------------------------------------------------
__global__ void kmeans_init_kernel(const float* __restrict__ cents_in,
                                   float* __restrict__ cents,
                                   float* __restrict__ npts,
                                   int* __restrict__ active) {
    int m = threadIdx.x;
    if (cents_in) {
        cents[m * 4 + 0] = cents_in[m * 3 + 0];
        cents[m * 4 + 1] = cents_in[m * 3 + 1];
        cents[m * 4 + 2] = cents_in[m * 3 + 2];
        cents[m * 4 + 3] = 0.0f;
    }
    npts[m] = 1.0f;
    if (m == 0) *active = 1;
}

// ---------------------------------------------------------------------------
// assign: WMMA-based argmax assignment + per-block cluster accumulation.
//
// sim-equivalent score per (point i, cluster j):
//   s = 2*x.c - |c|^2   (same argmax as reference's 2x.c - |x|^2 - |c|^2)
// computed as A(16x4) @ B(4x16) with A=[x,y,z,1], B col=[2cx,2cy,2cz,-|c|^2]
// via V_WMMA_F32_16X16X4_F32.
//
// Assumed fragment layouts (ISA 7.12.2):
//   A 16x4 f32 : lanes 0-15 -> M=lane, v0=K0, v1=K1 ; lanes 16-31 -> K2,K3
//   B 4x16 f32 : lanes 0-15 -> N=lane, v0=K0, v1=K1 ; lanes 16-31 -> K2,K3
//   D 16x16 f32: lane column N = lane%16; VGPR r = row r (+8 for lanes>=16)
// ---------------------------------------------------------------------------
__global__ void kmeans_assign_kernel(const float* __restrict__ pts,      // [N,3]
                                     const float* __restrict__ cents,    // [256*4]
                                     int*  __restrict__ closest,         // [N]
                                     float* __restrict__ partials,       // [NBLK][256][4]
                                     const int* __restrict__ active) {
    __shared__ float cb[NCLU][4];   // {2cx, 2cy, 2cz, -|c|^2}
    __shared__ float acc[NCLU][4];  // {count, sx, sy, sz}

    if (*active == 0) return;       // uniform across the whole block

    const int tid = threadIdx.x;
    {
        float cx = cents[tid * 4 + 0];
        float cy = cents[tid * 4 + 1];
        float cz = cents[tid * 4 + 2];
        cb[tid][0] = 2.0f * cx;
        cb[tid][1] = 2.0f * cy;
        cb[tid][2] = 2.0f * cz;
        cb[tid][3] = -(cx * cx + cy * cy + cz * cz);
        acc[tid][0] = 0.0f; acc[tid][1] = 0.0f;
        acc[tid][2] = 0.0f; acc[tid][3] = 0.0f;
    }
    __syncthreads();

    const int wave   = tid >> 5;
    const int lane   = tid & 31;
    const int lane16 = lane & 15;
    const bool hi    = (lane >= 16);

    for (int t = 0; t < TPW; ++t) {
        const int pbase = ((blockIdx.x * 8 + wave) * TPW + t) * 16;

        // ---- A fragment: 16 points, homogeneous coordinate ----
        v2f a;
        {
            const float* p = pts + (size_t)(pbase + lane16) * 3;
            if (!hi) { a.x = p[0]; a.y = p[1]; }     // K0=x, K1=y
            else     { a.x = p[2]; a.y = 1.0f; }     // K2=z, K3=1
        }

        float bestv[8];
        int   besti[8];
#pragma unroll
        for (int r = 0; r < 8; ++r) { bestv[r] = -3.4e38f; besti[r] = 0; }

        // ---- sweep 16 cluster tiles of 16 via WMMA ----
#pragma unroll 4
        for (int ct = 0; ct < 16; ++ct) {
            const int cl = ct * 16 + lane16;
            v2f bfrag;
            if (!hi) { bfrag.x = cb[cl][0]; bfrag.y = cb[cl][1]; }
            else     { bfrag.x = cb[cl][2]; bfrag.y = cb[cl][3]; }

            v8f c = {};
            c = __builtin_amdgcn_wmma_f32_16x16x4_f32(
                    false, a, false, bfrag, (short)0, c, false, false);

#pragma unroll
            for (int r = 0; r < 8; ++r) {
                float v = c[r];
                if (v > bestv[r]) { bestv[r] = v; besti[r] = cl; }  // strict > keeps first max
            }
        }

        // ---- butterfly argmax across the 16 lanes of each half-wave ----
#pragma unroll
        for (int msk = 1; msk <= 8; msk <<= 1) {
#pragma unroll
            for (int r = 0; r < 8; ++r) {
                float ov = __shfl_xor(bestv[r], msk, 32);
                int   oi = __shfl_xor(besti[r], msk, 32);
                if (ov > bestv[r] || (ov == bestv[r] && oi < besti[r])) {
                    bestv[r] = ov; besti[r] = oi;
                }
            }
        }

        // lanes 0-7 own rows 0-7, lanes 16-23 own rows 8-15
        int row = -1;
        if (lane < 8)                    row = lane;
        else if (lane >= 16 && lane < 24) row = 8 + (lane - 16);
        if (row >= 0) {
            const int cl = besti[row & 7];
            const int gi = pbase + row;
            closest[gi] = cl;
            const float* p = pts + (size_t)gi * 3;
            atomicAdd(&acc[cl][0], 1.0f);   // ds_add_f32
            atomicAdd(&acc[cl][1], p[0]);
            atomicAdd(&acc[cl][2], p[1]);
            atomicAdd(&acc[cl][3], p[2]);
        }
    }
    __syncthreads();

    float* dst = partials + ((size_t)blockIdx.x * NCLU + tid) * 4;
    dst[0] = acc[tid][0]; dst[1] = acc[tid][1];
    dst[2] = acc[tid][2]; dst[3] = acc[tid][3];
}

// ---------------------------------------------------------------------------
// update: deterministic partial reduction + centroid update + convergence flag
// ---------------------------------------------------------------------------
__global__ void kmeans_update_kernel(const float* __restrict__ partials,
                                     float* __restrict__ cents,
                                     float* __restrict__ npts,
                                     int* __restrict__ active) {
    if (*active == 0) return;
    const int m = threadIdx.x;

    float cnt = 0.f, sx = 0.f, sy = 0.f, sz = 0.f;
    for (int b = 0; b < NBLK; ++b) {                    // fixed order -> deterministic
        const float* p = partials + ((size_t)b * NCLU + m) * 4;
        cnt += p[0]; sx += p[1]; sy += p[2]; sz += p[3];
    }
    const float inv = 1.0f / (cnt + 1e-8f);
    const float gx = sx * inv, gy = sy * inv, gz = sz * inv;
    const float cx = cents[m * 4 + 0];
    const float cy = cents[m * 4 + 1];
    const float cz = cents[m * 4 + 2];
    const float dx = gx - cx, dy = gy - cy, dz = gz - cz;
    const float errm = dx * dx + dy * dy + dz * dz;

    const float np = npts[m];                           // npts BEFORE update
    const float lr = 0.9f / (np + 1e-8f) + 0.1f;
    cents[m * 4 + 0] = cx * (1.0f - lr) + gx * lr;
    cents[m * 4 + 1] = cy * (1.0f - lr) + gy * lr;
    cents[m * 4 + 2] = cz * (1.0f - lr) + gz * lr;
    npts[m] = np + cnt;

    __shared__ float red[NCLU];
    red[m] = errm;
    __syncthreads();
    for (int s = 128; s > 0; s >>= 1) {
        if (m < s) red[m] += red[m + s];
        __syncthreads();
    }
    if (m == 0 && red[0] <= TOLV) *active = 0;          // while(err > TOL)
}

// ---------------------------------------------------------------------------
// score: s_i = 2*x_i . c_{closest[i]} - |x_i|^2   (softmax/top-k invariant form)
// ---------------------------------------------------------------------------
__global__ void kmeans_score_kernel(const float* __restrict__ pts,
                                    const int* __restrict__ closest,
                                    const float* __restrict__ cents,
                                    float* __restrict__ score) {
    const int i = blockIdx.x * blockDim.x + threadIdx.x;
    const float* p = pts + (size_t)i * 3;
    const float x = p[0], y = p[1], z = p[2];
    const int cl = closest[i];
    const float cx = cents[cl * 4 + 0];
    const float cy = cents[cl * 4 + 1];
    const float cz = cents[cl * 4 + 2];
    score[i] = 2.0f * (x * cx + y * cy + z * cz) - (x * x + y * y + z * z);
}

// ---------------------------------------------------------------------------
// aggregate: one block per cluster. Register top-15 per thread (value desc,
// index asc on ties — matches lax.top_k stability), 256-way merge in LDS,
// softmax, weighted averages of points and features.
// ---------------------------------------------------------------------------
__global__ void kmeans_aggregate_kernel(const float* __restrict__ pts,     // [N,3]
                                        const float* __restrict__ fts,     // [F,N]
                                        const int*  __restrict__ closest,  // [N]
                                        const float* __restrict__ score,   // [N]
                                        float* __restrict__ out_cc,        // [256,3]
                                        float* __restrict__ out_cf) {      // [F,256]
    const int m   = blockIdx.x;
    const int tid = threadIdx.x;

    float lv[TOPK]; int li[TOPK];
#pragma unroll
    for (int k = 0; k < TOPK; ++k) { lv[k] = -3.4e38f; li[k] = 0x7fffffff; }

    for (int i = tid; i < N_PTS; i += 256) {
        float v = (closest[i] == m) ? score[i] : NEGV;
        int idx = i;
        if (v > lv[TOPK - 1] || (v == lv[TOPK - 1] && idx < li[TOPK - 1])) {
#pragma unroll
            for (int k = 0; k < TOPK; ++k) {
                if (v > lv[k] || (v == lv[k] && idx < li[k])) {
                    float tv = lv[k]; int ti = li[k];
                    lv[k] = v; li[k] = idx; v = tv; idx = ti;
                }
            }
        }
    }

    __shared__ float sv[256 * TOPK];
    __shared__ int   si[256 * TOPK];
    __shared__ int   head[256];
    __shared__ float redv[256];
    __shared__ int   redi[256];
    __shared__ int   redt[256];
    __shared__ float selv[TOPK];
    __shared__ int   seli[TOPK];
    __shared__ float w[TOPK];

#pragma unroll
    for (int k = 0; k < TOPK; ++k) { sv[tid * TOPK + k] = lv[k]; si[tid * TOPK + k] = li[k]; }
    head[tid] = 0;
    __syncthreads();

    // k-way merge: pick global best head 15 times
    for (int k = 0; k < TOPK; ++k) {
        const int h = head[tid];
        redv[tid] = (h < TOPK) ? sv[tid * TOPK + h] : -3.4e38f;
        redi[tid] = (h < TOPK) ? si[tid * TOPK + h] : 0x7fffffff;
        redt[tid] = tid;
        __syncthreads();
        for (int s = 128; s > 0; s >>= 1) {
            if (tid < s) {
                float ov = redv[tid + s]; int oi = redi[tid + s]; int ot = redt[tid + s];
                if (ov > redv[tid] || (ov == redv[tid] && oi < redi[tid])) {
                    redv[tid] = ov; redi[tid] = oi; redt[tid] = ot;
                }
            }
            __syncthreads();
        }
        if (tid == 0) { selv[k] = redv[0]; seli[k] = redi[0]; head[redt[0]]++; }
        __syncthreads();
    }

    // softmax over 15 scores (selv[0] is the max by construction)
    if (tid == 0) {
        const float mx = selv[0];
        float s = 0.f;
        for (int k = 0; k < TOPK; ++k) { float e = expf(selv[k] - mx); w[k] = e; s += e; }
        const float invs = 1.0f / s;
        for (int k = 0; k < TOPK; ++k) w[k] *= invs;
    }
    __syncthreads();

    if (tid < 3) {
        float s = 0.f;
#pragma unroll
        for (int k = 0; k < TOPK; ++k) s += w[k] * pts[(size_t)seli[k] * 3 + tid];
        out_cc[m * 3 + tid] = s;
    }
    {
        const float* row = fts + (size_t)tid * N_PTS;  // thread = feature index
        float s = 0.f;
#pragma unroll
        for (int k = 0; k < TOPK; ++k) s += w[k] * row[seli[k]];
        out_cf[tid * NCLU + m] = s;
    }
}

// ---------------------------------------------------------------------------
// final centroids -> d_out tail
// ---------------------------------------------------------------------------
__global__ void kmeans_final_kernel(const float* __restrict__ cents,
                                    float* __restrict__ out) {
    const int m = threadIdx.x;
    out[m * 3 + 0] = cents[m * 4 + 0];
    out[m * 3 + 1] = cents[m * 4 + 1];
    out[m * 3 + 2] = cents[m * 4 + 2];
}

// ---------------------------------------------------------------------------
extern "C" void kernel_launch(void* const* d_in, const int* in_sizes, int n_in,
                              void* d_out, int out_size, void* d_ws, size_t ws_size,
                              hipStream_t stream) {
    (void)in_sizes; (void)n_in; (void)out_size; (void)ws_size;

    const float* points     = (const float*)d_in[0];  // [2, 65536, 3]
    const float* features   = (const float*)d_in[1];  // [2, 256, 65536]
    const float* centroids0 = (const float*)d_in[2];  // [256, 3]

    float* out    = (float*)d_out;
    float* out_cc = out;                               // [2,256,3]
    float* out_cf = out + BZ * NCLU * 3;               // [2,256,256]
    float* out_fc = out_cf + BZ * NFEAT * NCLU;        // [256,3]

    char* ws = (char*)d_ws;
    float* cents    = (float*)ws;  ws += NCLU * 4 * sizeof(float);
    float* npts     = (float*)ws;  ws += NCLU * sizeof(float);
    int*   active   = (int*)ws;    ws += 256;  // padded
    float* partials = (float*)ws;  ws += (size_t)NBLK * NCLU * 4 * sizeof(float);
    int*   closest  = (int*)ws;    ws += (size_t)BZ * N_PTS * sizeof(int);
    float* score    = (float*)ws;  // N_PTS floats

    for (int b = 0; b < BZ; ++b) {
        // batch 0 loads input centroids; batch 1 carries them over (scan semantics)
        kmeans_init_kernel<<<1, 256, 0, stream>>>(
            (b == 0) ? centroids0 : (const float*)nullptr, cents, npts, active);

        const float* pb = points + (size_t)b * N_PTS * 3;
        int* cl_b = closest + (size_t)b * N_PTS;

        for (int it = 0; it < MAX_IT; ++it) {
            kmeans_assign_kernel<<<NBLK, 256, 0, stream>>>(pb, cents, cl_b, partials, active);
            kmeans_update_kernel<<<1, 256, 0, stream>>>(partials, cents, npts, active);
        }

        kmeans_score_kernel<<<N_PTS / 256, 256, 0, stream>>>(pb, cl_b, cents, score);
        kmeans_aggregate_kernel<<<NCLU, 256, 0, stream>>>(
            pb, features + (size_t)b * NFEAT * N_PTS, cl_b, score,
            out_cc + (size_t)b * NCLU * 3, out_cf + (size_t)b * NFEAT * NCLU);
    }

    kmeans_final_kernel<<<1, 256, 0, stream>>>(cents, out_fc);
}